// SoftDP_59571196396102
// MI455X (gfx1250) — compile-verified
//
#include <hip/hip_runtime.h>
#include <stdint.h>

// ---------------------------------------------------------------------------
// Soft value-iteration scan (backward in time) + softmax.
//   E: [B=2048, H=64, K=512] f32,  out: [B, K] f32
// One wave32 per batch row: V[k] register-resident (16 f32/lane),
// logsumexp via ds_swizzle XOR butterflies (1 DS op per stage, no barriers),
// E rows staged through a 4-deep LDS ring filled by CDNA5 async global->LDS
// DMA (ASYNCcnt), prefetch distance 3 time-steps.
// ---------------------------------------------------------------------------

namespace {
constexpr int B_ = 2048;
constexpr int H_ = 64;
constexpr int K_ = 512;
constexpr float KAPPA_ = 0.5f;
constexpr int WPB = 4;               // waves (batch rows) per block
constexpr int THREADS = WPB * 32;
constexpr int KPL = K_ / 32;         // 16 floats per lane
constexpr int NBUF = 4;              // LDS ring depth (prefetch distance 3)
}

// One butterfly stage: lane <- lane ^ XM, via ds_swizzle group-of-32 XOR mode.
// offset encoding: [14:10]=xor_mask, [9:5]=or_mask(0), [4:0]=and_mask(0x1F)
template <int XM>
__device__ __forceinline__ float swz_xor(float x) {
    return __int_as_float(
        __builtin_amdgcn_ds_swizzle(__float_as_int(x), (XM << 10) | 0x1F));
}

__device__ __forceinline__ float wave_max(float m) {
    m = fmaxf(m, swz_xor<16>(m));
    m = fmaxf(m, swz_xor<8>(m));
    m = fmaxf(m, swz_xor<4>(m));
    m = fmaxf(m, swz_xor<2>(m));
    m = fmaxf(m, swz_xor<1>(m));
    return m;
}

__device__ __forceinline__ float wave_sum(float s) {
    s += swz_xor<16>(s);
    s += swz_xor<8>(s);
    s += swz_xor<4>(s);
    s += swz_xor<2>(s);
    s += swz_xor<1>(s);
    return s;
}

__global__ __launch_bounds__(THREADS, 1)
void softdp_scan_kernel(const float* __restrict__ E, float* __restrict__ Out) {
    // Per-wave 4-deep ring: WPB waves * 4 bufs * 512 f32 = 32 KB
    __shared__ __align__(16) float smem[WPB * NBUF * K_];

    const int lane = threadIdx.x & 31;
    const int wave = threadIdx.x >> 5;
    const int b    = blockIdx.x * WPB + wave;

    const float* __restrict__ Erow = E + (size_t)b * (H_ * K_);

    // LDS byte address of this wave's staging area (low 32 bits of the
    // generic pointer are the LDS byte offset on gfx1250).
    const uint32_t lds_wave = (uint32_t)(uintptr_t)(&smem[wave * NBUF * K_]);
    const uint32_t lds_lane = lds_wave + (uint32_t)lane * 16u;

    // V[i] holds k = (i/4)*128 + lane*4 + (i%4)
    float V[KPL];
#pragma unroll
    for (int i = 0; i < KPL; ++i) V[i] = 0.0f;

    // Async-prefetch row t into ring slot `buf`: 4 x b128 per lane,
    // each instruction covers a contiguous 512B slab of the 2KB row.
    auto prefetch_row = [&](int t, int buf) {
        const char* g = (const char*)(Erow + (size_t)t * K_) + lane * 16;
        const uint32_t l = lds_lane + (uint32_t)buf * (K_ * 4);
#pragma unroll
        for (int j = 0; j < 4; ++j) {
            uint64_t ga = (uint64_t)(uintptr_t)(g + j * 512);
            uint32_t la = l + (uint32_t)(j * 512);
            asm volatile("global_load_async_to_lds_b128 %0, %1, off"
                         :: "v"(la), "v"(ga) : "memory");
        }
    };

    // Prologue: fill prefetch pipeline, distance 3 (12 DMAs outstanding).
    prefetch_row(H_ - 1, 0);
    prefetch_row(H_ - 2, 1);
    prefetch_row(H_ - 3, 2);

    for (int t = H_ - 1; t >= 0; --t) {
        const int buf = (H_ - 1 - t) & (NBUF - 1);

        if (t >= 3) {
            // Ring slot (buf+3)&3 was last ds_read one iteration ago; make
            // sure those reads drained before the DMA overwrites it.
            asm volatile("s_wait_dscnt 0x0" ::: "memory");
            prefetch_row(t - 3, (buf + 3) & (NBUF - 1));
            // <=12 outstanding -> the batch for the current slot has landed.
            asm volatile("s_wait_asynccnt 0xc" ::: "memory");
        } else if (t == 2) {
            asm volatile("s_wait_asynccnt 0x8" ::: "memory");
        } else if (t == 1) {
            asm volatile("s_wait_asynccnt 0x4" ::: "memory");
        } else {
            asm volatile("s_wait_asynccnt 0x0" ::: "memory");
        }

        // Pull this wave's staged 2KB row: 4 x ds_read_b128 per lane.
        const float4* sp = (const float4*)(smem + wave * NBUF * K_ + buf * K_);
        float4 e4[4];
#pragma unroll
        for (int j = 0; j < 4; ++j) e4[j] = sp[j * 32 + lane];
        const float* ef = (const float*)e4;   // ef[i] matches V[i]'s k

        // ---- wave-wide max of V ----
        float m = V[0];
#pragma unroll
        for (int i = 1; i < KPL; ++i) m = fmaxf(m, V[i]);
        m = wave_max(m);

        // ---- wave-wide sum of exp(V - m), keep ev for reuse ----
        float ev[KPL];
        float ssum = 0.0f;
#pragma unroll
        for (int i = 0; i < KPL; ++i) { ev[i] = __expf(V[i] - m); ssum += ev[i]; }
        ssum = wave_sum(ssum);

        // lse_switch = logsumexp(V) - kappa
        const float s  = m + __logf(ssum) - KAPPA_;
        const float mp = fmaxf(m, s);
        const float c  = __expf(s - mp);   // exp(s - mp)
        const float f  = __expf(m - mp);   // rescale: exp(V-m) -> exp(V-mp)

        // V_new = -e + logaddexp(s, V) = -e + mp + log(c + f*exp(V-m))
#pragma unroll
        for (int i = 0; i < KPL; ++i)
            V[i] = -ef[i] + mp + __logf(fmaf(f, ev[i], c));
    }

    // ---- final row softmax (tau = 1) ----
    float m2 = V[0];
#pragma unroll
    for (int i = 1; i < KPL; ++i) m2 = fmaxf(m2, V[i]);
    m2 = wave_max(m2);

    float p[KPL];
    float s2 = 0.0f;
#pragma unroll
    for (int i = 0; i < KPL; ++i) { p[i] = __expf(V[i] - m2); s2 += p[i]; }
    s2 = wave_sum(s2);

    const float inv = 1.0f / s2;
    float4* op = (float4*)(Out + (size_t)b * K_);
#pragma unroll
    for (int j = 0; j < 4; ++j) {
        float4 o;
        o.x = p[j * 4 + 0] * inv;
        o.y = p[j * 4 + 1] * inv;
        o.z = p[j * 4 + 2] * inv;
        o.w = p[j * 4 + 3] * inv;
        op[j * 32 + lane] = o;
    }
}

extern "C" void kernel_launch(void* const* d_in, const int* in_sizes, int n_in,
                              void* d_out, int out_size, void* d_ws, size_t ws_size,
                              hipStream_t stream) {
    (void)in_sizes; (void)n_in; (void)d_ws; (void)ws_size; (void)out_size;
    const float* E = (const float*)d_in[0];
    float* out = (float*)d_out;
    dim3 grid(B_ / WPB);
    dim3 block(THREADS);
    softdp_scan_kernel<<<grid, block, 0, stream>>>(E, out);
}